// TripletLoss_8323646619735
// MI455X (gfx1250) — compile-verified
//
#include <hip/hip_runtime.h>
#include <hip/hip_bf16.h>

typedef __attribute__((ext_vector_type(2))) float v2f;
typedef __attribute__((ext_vector_type(8))) float v8f;

#define B_ROWS 8192
#define DIMS   128
#define MARGIN 0.5f

// ---- order-preserving float <-> uint encoding for atomicMax ----
__device__ __forceinline__ unsigned int enc_f32(float f) {
    unsigned int u = __float_as_uint(f);
    return (u & 0x80000000u) ? ~u : (u | 0x80000000u);
}
__device__ __forceinline__ float dec_f32(unsigned int u) {
    unsigned int b = (u & 0x80000000u) ? (u ^ 0x80000000u) : ~u;
    return __uint_as_float(b);
}

// Pin 8 loaded fragments into live registers via a data dependence:
// consumers use the asm *outputs*, so they cannot be scheduled before the
// pin, and all 8 loads must be issued before the pin retires.
#define PIN8(a)                                                            \
    asm volatile("" : "+v"(a[0]), "+v"(a[1]), "+v"(a[2]), "+v"(a[3]),      \
                      "+v"(a[4]), "+v"(a[5]), "+v"(a[6]), "+v"(a[7]))

// ---------------- Kernel 1: normalize rows, compute pos_dist ----------------
// one wave (32 lanes) per row; 8 waves per 256-thread block
__global__ __launch_bounds__(256)
void normalize_kernel(const float* __restrict__ anchor,
                      const float* __restrict__ positive,
                      float* __restrict__ An,
                      float* __restrict__ posd) {
    const int warp = threadIdx.x >> 5;
    const int lane = threadIdx.x & 31;
    const int row  = blockIdx.x * 8 + warp;
    if (row >= B_ROWS) return;

    const float4 a4 = ((const float4*)(anchor   + (size_t)row * DIMS))[lane];
    const float4 p4 = ((const float4*)(positive + (size_t)row * DIMS))[lane];

    float sa = a4.x*a4.x + a4.y*a4.y + a4.z*a4.z + a4.w*a4.w;
    float sp = p4.x*p4.x + p4.y*p4.y + p4.z*p4.z + p4.w*p4.w;
    #pragma unroll
    for (int off = 1; off < 32; off <<= 1) {
        sa += __shfl_xor(sa, off, 32);
        sp += __shfl_xor(sp, off, 32);
    }
    const float ia = 1.0f / fmaxf(sqrtf(sa), 1e-12f);
    const float ip = 1.0f / fmaxf(sqrtf(sp), 1e-12f);

    float4 an = make_float4(a4.x*ia, a4.y*ia, a4.z*ia, a4.w*ia);
    float4 pn = make_float4(p4.x*ip, p4.y*ip, p4.z*ip, p4.w*ip);

    float dx = an.x-pn.x, dy = an.y-pn.y, dz = an.z-pn.z, dw = an.w-pn.w;
    float pd = dx*dx + dy*dy + dz*dz + dw*dw;
    #pragma unroll
    for (int off = 1; off < 32; off <<= 1) pd += __shfl_xor(pd, off, 32);

    ((float4*)(An + (size_t)row * DIMS))[lane] = an;
    if (lane == 0) posd[row] = pd;
}

// ---------------- Kernel 2: init encoded row-max buffer ----------------
__global__ __launch_bounds__(256)
void init_max_kernel(unsigned int* __restrict__ enc) {
    int i = blockIdx.x * 256 + threadIdx.x;
    if (i < B_ROWS) enc[i] = 0u;   // below any encoded real value
}

// ---------------- Kernel 3: WMMA gram + masked row-max ----------------
// block = 4 waves; wave w owns row tile i0 = (blockIdx.x*4+w)*16
// blockIdx.y selects a 1024-column chunk (64 tiles of 16).
// Two column tiles per iteration (independent accumulator chains cA/cB).
// B fragments are pinned into live registers (PIN8) so the backend must
// batch the loads into clauses with progressive s_wait_loadcnt instead of
// paying a full wait before every WMMA.
#define NCHUNK 8
#define COLS_PER_CHUNK  (B_ROWS / NCHUNK)      // 1024
#define TILES_PER_CHUNK (COLS_PER_CHUNK / 16)  // 64

__global__ __launch_bounds__(128)
void gram_max_kernel(const float* __restrict__ An,
                     const int* __restrict__ labels,
                     unsigned int* __restrict__ encmax) {
    const int warp = threadIdx.x >> 5;
    const int lane = threadIdx.x & 31;
    const int ln   = lane & 15;          // column-within-tile / row-within-tile
    const int hi   = lane >> 4;          // half-wave selector

    const int i0     = (blockIdx.x * 4 + warp) * 16;
    const int jstart = blockIdx.y * COLS_PER_CHUNK;

    // preload A fragments for all 32 k-steps: lane holds row (i0+ln), k = 4*kk + 2*hi
    v2f aF[32];
    const float* arow = An + (size_t)(i0 + ln) * DIMS + 2 * hi;
    #pragma unroll
    for (int kk = 0; kk < 32; ++kk)
        aF[kk] = *(const v2f*)(arow + 4 * kk);

    // labels for the 8 rows this lane contributes to (M = r + 8*hi)
    int labi[8];
    #pragma unroll
    for (int r = 0; r < 8; ++r)
        labi[r] = labels[i0 + r + 8 * hi];

    float rm[8];
    #pragma unroll
    for (int r = 0; r < 8; ++r) rm[r] = -INFINITY;

    for (int ct = 0; ct < TILES_PER_CHUNK; ct += 2) {
        const int colA  = jstart + ct * 16 + ln;
        const int colB  = colA + 16;
        const int labjA = labels[colA];
        const int labjB = labels[colB];
        const float* browA = An + (size_t)colA * DIMS + 2 * hi;
        const float* browB = An + (size_t)colB * DIMS + 2 * hi;

        // ---- load phase: 64 b64 loads into distinct registers ----
        v2f bA[32], bB[32];
        #pragma unroll
        for (int kk = 0; kk < 32; ++kk) {
            bA[kk] = *(const v2f*)(browA + 4 * kk);
            bB[kk] = *(const v2f*)(browB + 4 * kk);
        }

        // pin fragments in groups of 8: forces distinct live registers and
        // batched loads; later groups' loads may still hoist for overlap
        PIN8((&bA[0]));  PIN8((&bB[0]));
        PIN8((&bA[8]));  PIN8((&bB[8]));
        PIN8((&bA[16])); PIN8((&bB[16]));
        PIN8((&bA[24])); PIN8((&bB[24]));

        // ---- compute phase: 64 WMMAs, two interleaved chains ----
        v8f cA = {0.f,0.f,0.f,0.f,0.f,0.f,0.f,0.f};
        v8f cB = {0.f,0.f,0.f,0.f,0.f,0.f,0.f,0.f};
        #pragma unroll
        for (int kk = 0; kk < 32; ++kk) {
            cA = __builtin_amdgcn_wmma_f32_16x16x4_f32(
                     false, aF[kk], false, bA[kk], (short)0, cA, false, false);
            cB = __builtin_amdgcn_wmma_f32_16x16x4_f32(
                     false, aF[kk], false, bB[kk], (short)0, cB, false, false);
        }

        // mask same-label pairs, fold into running row max
        #pragma unroll
        for (int r = 0; r < 8; ++r) {
            float gA = cA[r];
            float gB = cB[r];
            rm[r] = (labi[r] != labjA) ? fmaxf(rm[r], gA) : rm[r];
            rm[r] = (labi[r] != labjB) ? fmaxf(rm[r], gB) : rm[r];
        }
    }

    // reduce across the 16 lanes of each half-wave (xor masks stay in-half)
    #pragma unroll
    for (int r = 0; r < 8; ++r) {
        float v = rm[r];
        #pragma unroll
        for (int off = 1; off < 16; off <<= 1)
            v = fmaxf(v, __shfl_xor(v, off, 32));
        rm[r] = v;
    }

    if (ln == 0) {
        #pragma unroll
        for (int r = 0; r < 8; ++r)
            atomicMax(&encmax[i0 + r + 8 * hi], enc_f32(rm[r]));
    }
}

// ---------------- Kernel 4: finalize mean loss ----------------
__global__ __launch_bounds__(256)
void finalize_kernel(const unsigned int* __restrict__ encmax,
                     const float* __restrict__ posd,
                     float* __restrict__ out) {
    __shared__ float sbuf[8];
    float sum = 0.0f;
    for (int i = threadIdx.x; i < B_ROWS; i += 256) {
        float maxg = dec_f32(encmax[i]);
        float neg  = fmaxf(2.0f - 2.0f * maxg, 0.0f);   // squared hard-negative dist
        float l    = fmaxf(posd[i] - neg + MARGIN, 0.0f);
        sum += l;
    }
    #pragma unroll
    for (int off = 1; off < 32; off <<= 1) sum += __shfl_xor(sum, off, 32);
    const int warp = threadIdx.x >> 5;
    const int lane = threadIdx.x & 31;
    if (lane == 0) sbuf[warp] = sum;
    __syncthreads();
    if (threadIdx.x == 0) {
        float t = 0.0f;
        #pragma unroll
        for (int w = 0; w < 8; ++w) t += sbuf[w];
        out[0] = t / (float)B_ROWS;
    }
}

extern "C" void kernel_launch(void* const* d_in, const int* in_sizes, int n_in,
                              void* d_out, int out_size, void* d_ws, size_t ws_size,
                              hipStream_t stream) {
    const float* anchor   = (const float*)d_in[0];
    const float* positive = (const float*)d_in[1];
    const int*   labels   = (const int*)d_in[2];
    float* out = (float*)d_out;

    float* An           = (float*)d_ws;                       // 8192*128 f32
    float* posd         = An + (size_t)B_ROWS * DIMS;         // 8192 f32
    unsigned int* encmx = (unsigned int*)(posd + B_ROWS);     // 8192 u32

    normalize_kernel<<<B_ROWS / 8, 256, 0, stream>>>(anchor, positive, An, posd);
    init_max_kernel<<<B_ROWS / 256, 256, 0, stream>>>(encmx);
    dim3 grid(B_ROWS / (4 * 16), NCHUNK);   // 128 x 8
    gram_max_kernel<<<grid, 128, 0, stream>>>(An, labels, encmx);
    finalize_kernel<<<1, 256, 0, stream>>>(encmx, posd, out);
}